// DDALoss_89129161327199
// MI455X (gfx1250) — compile-verified
//
#include <hip/hip_runtime.h>
#include <hip/hip_bf16.h>

typedef float v2f __attribute__((ext_vector_type(2)));
typedef float v8f __attribute__((ext_vector_type(8)));

#define NB            4096     // batch
#define DIM           512      // feature dim
#define NC            10000    // num classes
#define ROWS_PER_BLK  32
#define COLS_PER_IT   32
#define CHUNK         1024     // columns per grid-y chunk
#define NCHUNK        10       // ceil(10000/1024)
#define LDA           520      // padded LDS row stride (floats), 16B-aligned
#define LDB           520
#define NEG_BIG       (-1e30f)

// workspace layout (in floats)
#define WS_CS    0                     // [10000]  ||centers_c||^2
#define WS_T     10000                 // [4096]   target logit per row
#define WS_ACC   14096                 // [1]      centerloss accumulator
#define WS_PART  14112                 // [4096][NCHUNK*2][2] partial (max,sumexp)

// ---------------------------------------------------------------------------
// CDNA5 async global->LDS helpers (ASYNCcnt-tracked, bypasses VGPRs)
// ---------------------------------------------------------------------------
__device__ __forceinline__ unsigned __lds_byte_off(const void* p) {
    // __shared__ generic pointer -> 32-bit LDS byte offset
    return (unsigned)(uintptr_t)(const __attribute__((address_space(3))) void*)p;
}

__device__ __forceinline__ void async_g2l_b128(unsigned lds_byte,
                                               const float* gbase,
                                               int gbyte_off) {
    // GVS mode: global address is gbase (SGPR pair) plus gbyte_off (VGPR); 16 bytes go to LDS address in vdst
    asm volatile("global_load_async_to_lds_b128 %0, %1, %2"
                 :
                 : "v"(lds_byte), "v"(gbyte_off), "s"(gbase)
                 : "memory");
}

__device__ __forceinline__ void wait_async0() {
#if __has_builtin(__builtin_amdgcn_s_wait_asynccnt)
    __builtin_amdgcn_s_wait_asynccnt(0);
#else
    asm volatile("s_wait_asynccnt 0x0" ::: "memory");
#endif
}

// ---------------------------------------------------------------------------
// Kernel 1: cs[c] = sum_d centers[c,d]^2 ; also zero the centerloss accumulator
// ---------------------------------------------------------------------------
__global__ __launch_bounds__(256) void dda_prep(const float* __restrict__ centers,
                                                float* __restrict__ cs,
                                                float* __restrict__ accum) {
    const int wave = threadIdx.x >> 5;
    const int lane = threadIdx.x & 31;
    if (blockIdx.x == 0 && threadIdx.x == 0) *accum = 0.f;
    const int row = blockIdx.x * 8 + wave;
    if (row >= NC) return;
    const float4* p = (const float4*)(centers + (size_t)row * DIM);
    float s = 0.f;
#pragma unroll
    for (int i = 0; i < DIM / 4 / 32; ++i) {
        float4 v = p[lane + 32 * i];
        s += v.x * v.x + v.y * v.y + v.z * v.z + v.w * v.w;
    }
#pragma unroll
    for (int off = 16; off > 0; off >>= 1) s += __shfl_xor(s, off, 32);
    if (lane == 0) cs[row] = s;
}

// ---------------------------------------------------------------------------
// Kernel 2: per-row gather: centerloss partial + target logit t[n] = 2 f.c_l - cs[l]
// ---------------------------------------------------------------------------
__global__ __launch_bounds__(256) void dda_center(const float* __restrict__ feat,
                                                  const int* __restrict__ label,
                                                  const float* __restrict__ centers,
                                                  const float* __restrict__ cs,
                                                  float* __restrict__ tgt,
                                                  float* __restrict__ accum) {
    const int wave = threadIdx.x >> 5;
    const int lane = threadIdx.x & 31;
    const int n = blockIdx.x * 8 + wave;
    if (n >= NB) return;
    const int l = label[n];
    const float4* fp = (const float4*)(feat + (size_t)n * DIM);
    const float4* cp = (const float4*)(centers + (size_t)l * DIM);
    float dot = 0.f, sq = 0.f;
#pragma unroll
    for (int i = 0; i < DIM / 4 / 32; ++i) {
        float4 f = fp[lane + 32 * i];
        float4 c = cp[lane + 32 * i];
        float dx = f.x - c.x, dy = f.y - c.y, dz = f.z - c.z, dw = f.w - c.w;
        sq  += dx * dx + dy * dy + dz * dz + dw * dw;
        dot += f.x * c.x + f.y * c.y + f.z * c.z + f.w * c.w;
    }
#pragma unroll
    for (int off = 16; off > 0; off >>= 1) {
        dot += __shfl_xor(dot, off, 32);
        sq  += __shfl_xor(sq,  off, 32);
    }
    if (lane == 0) {
        tgt[n] = 2.f * dot - cs[l];
        atomicAdd(accum, sq);
    }
}

// ---------------------------------------------------------------------------
// Kernel 3: fused GEMM (f32 WMMA) + online logsumexp over column chunks.
//   logits s[n,c] = 2 * (feat_n . centers_c) - cs[c]   (||f||^2 cancels in NLL)
//   Each WG: 32 rows x one 1024-col chunk; 4 waves, each a 16x16 WMMA tile.
//   All LDS staging via global_load_async_to_lds_b128 (ASYNCcnt path).
// ---------------------------------------------------------------------------
__global__ __launch_bounds__(128) void dda_gemm_lse(const float* __restrict__ feat,
                                                    const float* __restrict__ centers,
                                                    const float* __restrict__ cs,
                                                    float* __restrict__ partial) {
    __shared__ float a_lds[ROWS_PER_BLK * LDA];   // 32 x 512 feat block (padded)
    __shared__ float b_lds[COLS_PER_IT * LDB];    // 32 x 512 centers tile (padded)

    const int tid    = threadIdx.x;
    const int wave   = tid >> 5;
    const int lane   = tid & 31;
    const int lane16 = lane & 15;
    const int hi     = lane >> 4;           // 0: K pair {k,k+1}, 1: {k+2,k+3}
    const int wave_m = wave >> 1;           // 0..1 -> rows
    const int wave_n = wave & 1;            // 0..1 -> cols
    const int row0     = blockIdx.x * ROWS_PER_BLK;
    const int col_base = blockIdx.y * CHUNK;

    const unsigned a_base = __lds_byte_off(a_lds);
    const unsigned b_base = __lds_byte_off(b_lds);

    // Stage the A row-block once (reused across all 1024 columns of this chunk)
    for (int t = tid; t < ROWS_PER_BLK * DIM / 4; t += 128) {
        int idx = t * 4;
        int r = idx >> 9;            // / DIM
        int k = idx & (DIM - 1);
        async_g2l_b128(a_base + (unsigned)(r * LDA + k) * 4u,
                       feat, ((row0 + r) * DIM + k) * 4);
    }

    float rmax[8], rsum[8];
#pragma unroll
    for (int j = 0; j < 8; ++j) { rmax[j] = NEG_BIG; rsum[j] = 0.f; }

    // per-lane fragment base pointers (ISA 16x4 A / 4x16 B f32 lane layout)
    const float* ap = a_lds + (wave_m * 16 + lane16) * LDA + 2 * hi;
    const float* bp = b_lds + (wave_n * 16 + lane16) * LDB + 2 * hi;

    for (int cb = col_base; cb < col_base + CHUNK; cb += COLS_PER_IT) {
        __syncthreads();    // previous iteration's b_lds readers done
        // Stage 32 center rows (columns of the logit matrix); zero-fill OOB tail
        for (int t = tid; t < COLS_PER_IT * DIM / 4; t += 128) {
            int idx = t * 4;
            int r = idx >> 9;
            int k = idx & (DIM - 1);
            int c = cb + r;
            if (c < NC) {
                async_g2l_b128(b_base + (unsigned)(r * LDB + k) * 4u,
                               centers, (c * DIM + k) * 4);
            } else {
                *(float4*)(b_lds + r * LDB + k) = make_float4(0.f, 0.f, 0.f, 0.f);
            }
        }
        wait_async0();      // own-wave async copies (A on first pass, B always) landed
        __syncthreads();    // all waves' copies visible

        // K-loop: two independent WMMA accumulation chains
        v8f acc0 = {}, acc1 = {};
#pragma unroll 4
        for (int k = 0; k < DIM; k += 8) {
            v2f a0 = *(const v2f*)(ap + k);
            v2f b0 = *(const v2f*)(bp + k);
            v2f a1 = *(const v2f*)(ap + k + 4);
            v2f b1 = *(const v2f*)(bp + k + 4);
            acc0 = __builtin_amdgcn_wmma_f32_16x16x4_f32(
                false, a0, false, b0, (short)0, acc0, false, false);
            acc1 = __builtin_amdgcn_wmma_f32_16x16x4_f32(
                false, a1, false, b1, (short)0, acc1, false, false);
        }

        const int  col   = cb + wave_n * 16 + lane16;
        const bool valid = col < NC;
        const float csv  = valid ? cs[col] : 0.f;

        // online logsumexp update; row M = wave_m*16 + 8*hi + j, 16 lanes = 16 cols
#pragma unroll
        for (int j = 0; j < 8; ++j) {
            float s = valid ? fmaf(2.f, acc0[j] + acc1[j], -csv) : NEG_BIG;
            float tmax = s;
            tmax = fmaxf(tmax, __shfl_xor(tmax, 1, 16));
            tmax = fmaxf(tmax, __shfl_xor(tmax, 2, 16));
            tmax = fmaxf(tmax, __shfl_xor(tmax, 4, 16));
            tmax = fmaxf(tmax, __shfl_xor(tmax, 8, 16));
            float mnew = fmaxf(rmax[j], tmax);
            float e = __expf(s - mnew);
            e += __shfl_xor(e, 1, 16);
            e += __shfl_xor(e, 2, 16);
            e += __shfl_xor(e, 4, 16);
            e += __shfl_xor(e, 8, 16);
            rsum[j] = rsum[j] * __expf(rmax[j] - mnew) + e;
            rmax[j] = mnew;
        }
    }

    // lanes 0 and 16 hold fully-reduced state for their 8 rows each
    if (lane16 == 0) {
#pragma unroll
        for (int j = 0; j < 8; ++j) {
            int row = row0 + wave_m * 16 + 8 * hi + j;
            int pi = (row * (NCHUNK * 2) + (int)blockIdx.y * 2 + wave_n) * 2;
            partial[pi]     = rmax[j];
            partial[pi + 1] = rsum[j];
        }
    }
}

// ---------------------------------------------------------------------------
// Kernel 4: merge partials -> lse per row -> nll -> final 3 scalars
// ---------------------------------------------------------------------------
__global__ __launch_bounds__(256) void dda_final(const float* __restrict__ partial,
                                                 const float* __restrict__ tgt,
                                                 const float* __restrict__ accum,
                                                 float* __restrict__ out) {
    __shared__ float red[256];
    float local = 0.f;
    for (int n = threadIdx.x; n < NB; n += 256) {
        const float* p = partial + (size_t)n * (NCHUNK * 2) * 2;
        float M = NEG_BIG, S = 0.f;
#pragma unroll
        for (int i = 0; i < NCHUNK * 2; ++i) {
            float m = p[2 * i], s = p[2 * i + 1];
            if (m > M) { S = S * __expf(M - m) + s; M = m; }
            else       { S += s * __expf(m - M); }
        }
        float lse = M + logf(S);
        local += lse - tgt[n];
    }
    red[threadIdx.x] = local;
    __syncthreads();
    for (int s2 = 128; s2 > 0; s2 >>= 1) {
        if (threadIdx.x < s2) red[threadIdx.x] += red[threadIdx.x + s2];
        __syncthreads();
    }
    if (threadIdx.x == 0) {
        float nll = red[0] / (float)NB;              // mean over batch
        float dda = nll / (float)NB / 2.f;           // / N / 2
        float cl  = (*accum) / 2.f / (float)NB;      // / 2 / N
        out[0] = 0.01f * cl + 3.0f * dda;            // LAMB, GAMMA
        out[1] = cl;
        out[2] = dda;
    }
}

// ---------------------------------------------------------------------------
extern "C" void kernel_launch(void* const* d_in, const int* in_sizes, int n_in,
                              void* d_out, int out_size, void* d_ws, size_t ws_size,
                              hipStream_t stream) {
    const float* feat    = (const float*)d_in[0];
    const int*   label   = (const int*)d_in[1];
    const float* centers = (const float*)d_in[2];

    float* ws    = (float*)d_ws;
    float* cs    = ws + WS_CS;
    float* tgt   = ws + WS_T;
    float* accum = ws + WS_ACC;
    float* part  = ws + WS_PART;

    dda_prep<<<(NC + 7) / 8, 256, 0, stream>>>(centers, cs, accum);
    dda_center<<<NB / 8, 256, 0, stream>>>(feat, label, centers, cs, tgt, accum);
    dda_gemm_lse<<<dim3(NB / ROWS_PER_BLK, NCHUNK), 128, 0, stream>>>(feat, centers, cs, part);
    dda_final<<<1, 256, 0, stream>>>(part, tgt, accum, (float*)d_out);
}